// GraphConstruct_spatial_74285754351629
// MI455X (gfx1250) — compile-verified
//
#include <hip/hip_runtime.h>
#include <hip/hip_bf16.h>
#include <float.h>

typedef __attribute__((ext_vector_type(2))) float v2f;
typedef __attribute__((ext_vector_type(8))) float v8f;

#define Mn 8192
#define En 256
#define Kn 16
#define CHUNK 512
#define DSTR 513                 // 16 x 512 d2 tile, padded row stride (bank spread)

#define SCORE_BASE ((size_t)0)
#define IDX_BASE   ((size_t)262144)          // 16*2*8192
#define DIFF_BASE  ((size_t)393216)          // + 8192*16

// ---------------------------------------------------------------------------
// Kernel 1: fused pairwise-distance (via V_WMMA_F32_16X16X4_F32) + top-K.
// d2(i,j) = |pi|^2 + |pj|^2 - 2 pi.pj  folded into one K=4 WMMA dot product:
//   A row i   = [-2xi, -2yi, |pi|^2, 1 ]
//   B col j   = [  xj,   yj,   1, |pj|^2]
// Chunk staged in LDS pre-swizzled into B layout (one ds_load_b64 per lane).
// Top-K selection runs on clamped d2 (sqrt is monotonic); sqrt applied only
// to the final 16 values per row at output time.
// ---------------------------------------------------------------------------
__global__ __launch_bounds__(256) void knn_wmma_kernel(const float* __restrict__ spatial,
                                                       float* __restrict__ out) {
    __shared__ v2f   s_bv[2 * CHUNK];    // B-format staging (8 KB)
    __shared__ float s_buf[16 * DSTR];   // d2 tile; reused as merge buffer at the end

    const int tid     = threadIdx.x;
    const int lane    = tid & 31;        // wave32
    const int half    = lane >> 4;       // lane half selects K-pair in A/B layout
    const int lmod    = lane & 15;
    const int wave    = tid >> 5;        // 8 waves
    const int rowBase = blockIdx.x * 16;

    // --- A operand (identical for every wave of this block; built once) ---
    // lanes 0-15 : VGPR0=K0, VGPR1=K1 ; lanes 16-31 : VGPR0=K2, VGPR1=K3
    float px = spatial[2 * (rowBase + lmod) + 0];
    float py = spatial[2 * (rowBase + lmod) + 1];
    v2f A;
    if (half == 0) { A.x = -2.0f * px;        A.y = -2.0f * py; }
    else           { A.x = px * px + py * py; A.y = 1.0f;       }

    // --- per-thread sorted top-K (ascending d2), register resident ---
    float bd[Kn]; int bi[Kn];
#pragma unroll
    for (int i = 0; i < Kn; ++i) { bd[i] = FLT_MAX; bi[i] = 0x7fffffff; }

    const int selRow  = tid & 15;        // 16 threads per query row
    const int selSlot = tid >> 4;        // each scans 32 columns of the chunk

    for (int cb = 0; cb < Mn; cb += CHUNK) {
        // stage candidate chunk in B-matrix layout
        for (int c = tid; c < CHUNK; c += 256) {
            float x = spatial[2 * (cb + c) + 0];
            float y = spatial[2 * (cb + c) + 1];
            v2f lo; lo.x = x;    lo.y = y;
            v2f hi; hi.x = 1.0f; hi.y = x * x + y * y;
            s_bv[c]         = lo;
            s_bv[CHUNK + c] = hi;
        }
        __syncthreads();

        // 8 waves x 4 WMMA tiles = 512 columns of d2 into LDS
#pragma unroll
        for (int sub = 0; sub < 4; ++sub) {
            const int cl = sub * 128 + wave * 16 + lmod;
            v2f B = s_bv[half * CHUNK + cl];          // one ds_load_b64, no branches
            v8f C = {0.f, 0.f, 0.f, 0.f, 0.f, 0.f, 0.f, 0.f};
            C = __builtin_amdgcn_wmma_f32_16x16x4_f32(
                    /*neg_a=*/false, A, /*neg_b=*/false, B,
                    /*c_mod=*/(short)0, C, /*reuse_a=*/false, /*reuse_b=*/false);
            // D layout: VGPR r -> row r (lanes 0-15) / row r+8 (lanes 16-31), col = lane&15
            const int rrow = half * 8;
#pragma unroll
            for (int r = 0; r < 8; ++r)
                s_buf[(rrow + r) * DSTR + cl] = C[r];
        }
        __syncthreads();

        // selection on clamped d2 (sqrt deferred): insert 32 candidates
#pragma unroll 1
        for (int i = 0; i < 32; ++i) {
            const int c  = selSlot * 32 + i;
            const float d = fmaxf(s_buf[selRow * DSTR + c], 0.0f);   // matches ref clamp
            const int   j = cb + c;
            bool ins = (d < bd[Kn - 1]) || (d == bd[Kn - 1] && j < bi[Kn - 1]);
            if (ins) {
                bd[Kn - 1] = d; bi[Kn - 1] = j;
#pragma unroll
                for (int p = Kn - 1; p >= 1; --p) {
                    bool sw = (bd[p] < bd[p - 1]) ||
                              (bd[p] == bd[p - 1] && bi[p] < bi[p - 1]);
                    if (sw) {
                        float td = bd[p]; bd[p] = bd[p - 1]; bd[p - 1] = td;
                        int   ti = bi[p]; bi[p] = bi[p - 1]; bi[p - 1] = ti;
                    }
                }
            }
        }
        __syncthreads();
    }

    // --- merge: dump 16 sorted sublists per row into LDS (reuse s_buf) ---
    float* mscore = s_buf;                          // 16*256 floats
    int*   midx   = ((int*)s_buf) + 16 * 256;       // 16*256 ints (8192 <= 8208 slots)
#pragma unroll
    for (int e = 0; e < Kn; ++e) {
        mscore[selRow * 256 + selSlot * Kn + e] = bd[e];
        midx  [selRow * 256 + selSlot * Kn + e] = bi[e];
    }
    __syncthreads();

    if (tid < 16) {
        const int row = tid;
        const int mm  = rowBase + row;
        for (int e = 0; e < Kn; ++e) {
            float best = FLT_MAX; int bestj = 0x7fffffff; int bestc = 0;
            for (int c = 0; c < 256; ++c) {
                float d = mscore[row * 256 + c];
                int   j = midx  [row * 256 + c];
                if (d < best || (d == best && j < bestj)) { best = d; bestj = j; bestc = c; }
            }
            mscore[row * 256 + bestc] = FLT_MAX;    // consume

            out[IDX_BASE + (size_t)mm * Kn + e] = (float)bestj;   // idx_k
            float sc = __expf(-sqrtf(best) * 0.1f);                // score = exp(-d/10)
            size_t sb = SCORE_BASE + (size_t)e * (2 * Mn) + mm;
            out[sb]      = sc;                                     // scale copy 0
            out[sb + Mn] = sc;                                     // scale copy 1
        }
    }
}

// ---------------------------------------------------------------------------
// Kernel 2: diff_patch[k*E+e, s*M+mm] = |ye[mm,e] - xe[idx[mm,k],e]|
// mm is the fastest thread dimension -> the 268 MB of stores are coalesced;
// xe/ye (8 MB each) live in the 192 MB L2, so the gather reads are cheap.
// ---------------------------------------------------------------------------
__global__ __launch_bounds__(256) void diff_patch_kernel(const float* __restrict__ xe,
                                                         const float* __restrict__ ye,
                                                         float* __restrict__ out) {
    const int mm    = blockIdx.x * 256 + threadIdx.x;
    const int k     = blockIdx.z;
    const int ebase = blockIdx.y * 4;
    const int i = (int)out[IDX_BASE + (size_t)mm * Kn + k];  // written by kernel 1
#pragma unroll
    for (int e4 = 0; e4 < 4; ++e4) {
        const int e = ebase + e4;
        float v = fabsf(ye[(size_t)mm * En + e] - xe[(size_t)i * En + e]);
        size_t base = DIFF_BASE + ((size_t)(k * En + e) * 2) * Mn + mm;
        out[base]      = v;   // s = 0
        out[base + Mn] = v;   // s = 1
    }
}

extern "C" void kernel_launch(void* const* d_in, const int* in_sizes, int n_in,
                              void* d_out, int out_size, void* d_ws, size_t ws_size,
                              hipStream_t stream) {
    (void)in_sizes; (void)n_in; (void)out_size; (void)d_ws; (void)ws_size;
    const float* xe      = (const float*)d_in[0];
    const float* ye      = (const float*)d_in[1];
    const float* spatial = (const float*)d_in[2];
    float* out = (float*)d_out;

    knn_wmma_kernel<<<dim3(Mn / 16), 256, 0, stream>>>(spatial, out);
    diff_patch_kernel<<<dim3(Mn / 256, En / 4, Kn), 256, 0, stream>>>(xe, ye, out);
}